// GNNEncoderFullVariable_88502096101410
// MI455X (gfx1250) — compile-verified
//
#include <hip/hip_runtime.h>

#define NB    128
#define LMAX  6
#define PW    64
#define PB    32
#define POWW  256
#define POB   64
#define T2    1024
#define ENCIN 1056
#define HID   512
#define HR    256
#define OUTD  512

typedef float v2f __attribute__((ext_vector_type(2)));
typedef float v8f __attribute__((ext_vector_type(8)));

// ---------------------------------------------------------------------------
// Kernel 1: scrambled lengths (pack_padded_sequence reproduction). One block,
// 128 threads (one per batch).
// ---------------------------------------------------------------------------
__global__ void k_lens(const int* __restrict__ glen_g,
                       const int* __restrict__ Awp, const int* __restrict__ Abp,
                       const int* __restrict__ Vwp, const int* __restrict__ Vbp,
                       int* __restrict__ lwA, int* __restrict__ lbA,
                       int* __restrict__ lwV, int* __restrict__ lbV,
                       int* __restrict__ valid)
{
    __shared__ int glen[NB];
    __shared__ int offs[LMAX];
    __shared__ int flatAw[NB*LMAX + 1], flatAb[NB*LMAX + 1];
    __shared__ int flatVw[NB*LMAX + 1], flatVb[NB*LMAX + 1];

    const int b = threadIdx.x;
    if (b < NB) glen[b] = glen_g[b];
    __syncthreads();

    for (int i = threadIdx.x; i <= NB*LMAX; i += blockDim.x) {
        flatAw[i] = 0; flatAb[i] = 0; flatVw[i] = 0; flatVb[i] = 0;
    }
    if (threadIdx.x < LMAX) {   // offset[l] = sum_b min(glen[b], l)
        int l = threadIdx.x, o = 0;
        for (int j = 0; j < NB; ++j) o += (glen[j] < l) ? glen[j] : l;
        offs[l] = o;
    }
    int r = 0, st = 0;
    if (b < NB) {               // stable descending rank + batch-major start
        const int g = glen[b];
        for (int j = 0; j < NB; ++j) {
            const int gj = glen[j];
            r  += (gj > g) || (gj == g && j < b);
            st += (j < b) ? gj : 0;
        }
    }
    __syncthreads();
    if (b < NB) {               // scatter plen into packed-flat arrays
        const int g = glen[b];
        const int aw = Awp[b], ab = Abp[b], vw = Vwp[b], vb = Vbp[b];
        for (int l = 0; l < g; ++l) {
            flatAw[st + l] = aw; flatAb[st + l] = ab;
            flatVw[st + l] = vw; flatVb[st + l] = vb;
        }
    }
    __syncthreads();
    if (b < NB) {
        const int g = glen[b];
        for (int l = 0; l < LMAX; ++l) {
            const int v   = (l < g) ? 1 : 0;
            const int idx = b*LMAX + l;
            int i = offs[l] + r; if (i > NB*LMAX) i = NB*LMAX;
            valid[idx] = v;
            lwA[idx] = v ? flatAw[i] : 0;
            lbA[idx] = v ? flatAb[i] : 0;
            lwV[idx] = v ? flatVw[i] : 0;
            lbV[idx] = v ? flatVb[i] : 0;
        }
    }
}

// ---------------------------------------------------------------------------
// Kernel 2: masked prefix sum per row: S[r] = sum_{t<len[r]} x[r,t]
// ---------------------------------------------------------------------------
__global__ void k_rowsum(const float* __restrict__ x, const int* __restrict__ lens,
                         int P, int nrows, float* __restrict__ S)
{
    const int r = blockIdx.x * blockDim.x + threadIdx.x;
    if (r >= nrows) return;
    int L = lens[r]; if (L > P) L = P; if (L < 0) L = 0;
    const float* row = x + (size_t)r * P;
    float s = 0.f;
    for (int t = 0; t < L; ++t) s += row[t];
    S[r] = s;
}

// ---------------------------------------------------------------------------
// Kernel 3: emb[r,d] = sigmoid(S*w[d] + len*b[d])  (rank-1 collapsed form)
// ---------------------------------------------------------------------------
__global__ void k_emb(const float* __restrict__ Sw, const int* __restrict__ lw,
                      const float* __restrict__ Sb, const int* __restrict__ lb,
                      const float* __restrict__ wt_w, const float* __restrict__ wt_b,
                      const float* __restrict__ bt_w, const float* __restrict__ bt_b,
                      float* __restrict__ emb, int nrows)
{
    const int idx = blockIdx.x * blockDim.x + threadIdx.x;
    const int total = nrows * ENCIN;
    if (idx >= total) return;
    const int r = idx / ENCIN, d = idx - r * ENCIN;
    float x;
    if (d < T2) x = Sw[r] * wt_w[d] + (float)lw[r] * wt_b[d];
    else { const int d2 = d - T2; x = Sb[r] * bt_w[d2] + (float)lb[r] * bt_b[d2]; }
    emb[idx] = 1.f / (1.f + expf(-x));
}

// ---------------------------------------------------------------------------
// Kernel 4: universal WMMA GEMM, register-blocked BMx(BN) 16x16 subtiles.
//   C[m,n] = post( sum_k A1[m,k]W1[n,k] + sum_k A2[m,k]W2[n,k] + b1 + b2 )
// post: act (0=none,1=relu,2=tanh), optional +Cadd, optional valid-select.
// V_WMMA_F32_16X16X4_F32, K stepped by 4.
// A frag: row = lane&15, K-pair = lane>>4 (ISA 16x4 layout).
// B frag: W^T[k,n] -> b = {W[n, k+2h], W[n, k+2h+1]}, n = lane&15.
// D: m = v + 8*(lane>>4), n = lane&15.
// BMxBN blocking: each A frag feeds BN wmmas, each B frag feeds BM wmmas
// (halves L2 traffic per wmma at 2x2 vs 1x1).
// ---------------------------------------------------------------------------
template<int BM, int BN>
__global__ __launch_bounds__(256)
void k_gemm(const float* __restrict__ A1, int lda1, int K1,
            const float* __restrict__ W1, int ldw1,
            const float* __restrict__ A2, int lda2, int K2,
            const float* __restrict__ W2, int ldw2,
            const float* __restrict__ bias1, const float* __restrict__ bias2,
            const float* __restrict__ Cadd,
            const int* __restrict__ valid, int vstride,
            const float* __restrict__ Hkeep,
            float* __restrict__ C, int ldc, int M, int N, int act)
{
    const int wave = (blockIdx.x * blockDim.x + threadIdx.x) >> 5;
    const int lane = threadIdx.x & 31;
    const int nt_n = N / (16 * BN);
    const int ntiles = (M / (16 * BM)) * nt_n;
    if (wave >= ntiles) return;           // wave-uniform: EXEC stays all-ones
    const int mt = wave / nt_n, nt = wave - mt * nt_n;
    const int m0 = mt * 16 * BM, n0 = nt * 16 * BN;
    const int half = lane >> 4;
    const int r16  = lane & 15;

    v8f acc[BM][BN];
    #pragma unroll
    for (int i = 0; i < BM; ++i)
        #pragma unroll
        for (int j = 0; j < BN; ++j)
            acc[i][j] = (v8f){0.f,0.f,0.f,0.f,0.f,0.f,0.f,0.f};

    {
        const float* Ar = A1 + (size_t)(m0 + r16) * lda1 + 2*half;
        const float* Br = W1 + (size_t)(n0 + r16) * ldw1 + 2*half;
        #pragma unroll 4
        for (int k = 0; k < K1; k += 4) {
            v2f a[BM], b[BN];
            #pragma unroll
            for (int i = 0; i < BM; ++i) a[i] = *(const v2f*)(Ar + (size_t)i*16*lda1 + k);
            #pragma unroll
            for (int j = 0; j < BN; ++j) b[j] = *(const v2f*)(Br + (size_t)j*16*ldw1 + k);
            #pragma unroll
            for (int i = 0; i < BM; ++i)
                #pragma unroll
                for (int j = 0; j < BN; ++j)
                    acc[i][j] = __builtin_amdgcn_wmma_f32_16x16x4_f32(
                        false, a[i], false, b[j], (short)0, acc[i][j], false, false);
        }
    }
    if (A2) {
        const float* Ar = A2 + (size_t)(m0 + r16) * lda2 + 2*half;
        const float* Br = W2 + (size_t)(n0 + r16) * ldw2 + 2*half;
        #pragma unroll 4
        for (int k = 0; k < K2; k += 4) {
            v2f a[BM], b[BN];
            #pragma unroll
            for (int i = 0; i < BM; ++i) a[i] = *(const v2f*)(Ar + (size_t)i*16*lda2 + k);
            #pragma unroll
            for (int j = 0; j < BN; ++j) b[j] = *(const v2f*)(Br + (size_t)j*16*ldw2 + k);
            #pragma unroll
            for (int i = 0; i < BM; ++i)
                #pragma unroll
                for (int j = 0; j < BN; ++j)
                    acc[i][j] = __builtin_amdgcn_wmma_f32_16x16x4_f32(
                        false, a[i], false, b[j], (short)0, acc[i][j], false, false);
        }
    }

    #pragma unroll
    for (int j = 0; j < BN; ++j) {
        const int n = n0 + j*16 + r16;
        const float bv = (bias1 ? bias1[n] : 0.f) + (bias2 ? bias2[n] : 0.f);
        #pragma unroll
        for (int i = 0; i < BM; ++i) {
            #pragma unroll
            for (int v = 0; v < 8; ++v) {
                const int m = m0 + i*16 + v + 8*half;
                float x = acc[i][j][v] + bv;
                if (act == 1)      x = fmaxf(x, 0.f);
                else if (act == 2) x = tanhf(x);
                if (Cadd)  x += Cadd[(size_t)m * ldc + n];
                if (valid) {
                    const float keep = Hkeep[(size_t)m * ldc + n];
                    x = valid[m * vstride] ? x : keep;
                }
                C[(size_t)m * ldc + n] = x;
            }
        }
    }
}

// ---------------------------------------------------------------------------
// Host side
// ---------------------------------------------------------------------------
enum {
    P_IN_WT_W, P_IN_WT_B, P_IN_BT_W, P_IN_BT_B, P_IN_ENC_W, P_IN_ENC_B,
    P_OUT_WT_W, P_OUT_WT_B, P_OUT_BT_W, P_OUT_BT_B, P_OUT_ENC_W, P_OUT_ENC_B,
    P_A_WT_W, P_A_WT_B, P_A_BT_W, P_A_BT_B, P_A_ENC_W, P_A_ENC_B,
    P_V_WT_W, P_V_WT_B, P_V_BT_W, P_V_BT_B, P_V_ENC_W, P_V_ENC_B,
    P_RNN_WIH, P_RNN_WHH, P_RNN_BIH, P_RNN_BHH, P_SI_W, P_SI_B, P_OUTP_W, P_OUTP_B,
    P_COUNT
};
static const int p_sizes[P_COUNT] = {
    1024,1024,32,32,540672,512,  1024,1024,32,32,540672,512,
    1024,1024,32,32,540672,512,  1024,1024,32,32,540672,512,
    131072,65536,256,256,131072,256,393216,512
};
// alphabetical leaf order -> canonical index (if harness jax-sorts dict keys)
static const int alpha_to_canon[P_COUNT] = {
    P_A_BT_B, P_A_BT_W, P_A_ENC_B, P_A_ENC_W, P_A_WT_B, P_A_WT_W,
    P_V_BT_B, P_V_BT_W, P_V_ENC_B, P_V_ENC_W, P_V_WT_B, P_V_WT_W,
    P_IN_BT_B, P_IN_BT_W, P_IN_ENC_B, P_IN_ENC_W, P_IN_WT_B, P_IN_WT_W,
    P_OUT_BT_B, P_OUT_BT_W, P_OUT_ENC_B, P_OUT_ENC_W, P_OUT_WT_B, P_OUT_WT_W,
    P_OUTP_B, P_OUTP_W, P_RNN_BHH, P_RNN_BIH, P_RNN_WHH, P_RNN_WIH, P_SI_B, P_SI_W
};

static void launch_gemm(hipStream_t s, bool big,
                        const float* A1, int lda1, int K1, const float* W1, int ldw1,
                        const float* A2, int lda2, int K2, const float* W2, int ldw2,
                        const float* b1, const float* b2, const float* Cadd,
                        const int* valid, int vstride, const float* Hkeep,
                        float* C, int ldc, int M, int N, int act)
{
    if (big) {
        const int tiles = (M / 32) * (N / 32);
        const int blocks = (tiles + 7) / 8;     // 8 waves per 256-thread block
        k_gemm<2,2><<<blocks, 256, 0, s>>>(A1, lda1, K1, W1, ldw1, A2, lda2, K2, W2, ldw2,
                                           b1, b2, Cadd, valid, vstride, Hkeep,
                                           C, ldc, M, N, act);
    } else {
        const int tiles = (M / 16) * (N / 16);
        const int blocks = (tiles + 7) / 8;
        k_gemm<1,1><<<blocks, 256, 0, s>>>(A1, lda1, K1, W1, ldw1, A2, lda2, K2, W2, ldw2,
                                           b1, b2, Cadd, valid, vstride, Hkeep,
                                           C, ldc, M, N, act);
    }
}

extern "C" void kernel_launch(void* const* d_in, const int* in_sizes, int n_in,
                              void* d_out, int out_size, void* d_ws, size_t ws_size,
                              hipStream_t stream)
{
    (void)out_size; (void)ws_size;

    // ---- top-level inputs (insertion vs alphabetical order, by size probe) --
    const float *A_weight,*A_bias,*V_weight,*V_bias,*out_weight,*out_bias,*in_weight,*in_bias;
    const int *gnn,*Awp,*Abp,*Vwp,*Vbp,*Owp,*Obp,*Iwp,*Ibp;
    if (in_sizes[0] == NB*LMAX*PW) {               // insertion order
        A_weight=(const float*)d_in[0];  A_bias=(const float*)d_in[1];
        V_weight=(const float*)d_in[2];  V_bias=(const float*)d_in[3];
        out_weight=(const float*)d_in[4]; out_bias=(const float*)d_in[5];
        in_weight=(const float*)d_in[6]; in_bias=(const float*)d_in[7];
        gnn=(const int*)d_in[8];
        Awp=(const int*)d_in[9];  Abp=(const int*)d_in[10];
        Vwp=(const int*)d_in[11]; Vbp=(const int*)d_in[12];
        Owp=(const int*)d_in[13]; Obp=(const int*)d_in[14];
        Iwp=(const int*)d_in[15]; Ibp=(const int*)d_in[16];
    } else {                                        // alphabetical order
        A_bias=(const float*)d_in[0];  Abp=(const int*)d_in[1];
        A_weight=(const float*)d_in[2]; Awp=(const int*)d_in[3];
        V_bias=(const float*)d_in[4];  Vbp=(const int*)d_in[5];
        V_weight=(const float*)d_in[6]; Vwp=(const int*)d_in[7];
        gnn=(const int*)d_in[8];
        in_bias=(const float*)d_in[9];  Ibp=(const int*)d_in[10];
        in_weight=(const float*)d_in[11]; Iwp=(const int*)d_in[12];
        out_bias=(const float*)d_in[13]; Obp=(const int*)d_in[14];
        out_weight=(const float*)d_in[15]; Owp=(const int*)d_in[16];
    }

    // ---- params leaves ------------------------------------------------------
    const float* P[P_COUNT];
    const int pstart = 17;
    if (n_in <= pstart + 1) {                      // single concatenated buffer
        const float* base = (const float*)d_in[pstart];
        size_t off = 0;
        for (int i = 0; i < P_COUNT; ++i) { P[i] = base + off; off += p_sizes[i]; }
    } else if (in_sizes[pstart] == 1024) {         // insertion-order leaves
        for (int i = 0; i < P_COUNT; ++i) P[i] = (const float*)d_in[pstart + i];
    } else {                                       // alphabetical leaves
        for (int i = 0; i < P_COUNT; ++i) P[alpha_to_canon[i]] = (const float*)d_in[pstart + i];
    }

    // ---- scratch layout -----------------------------------------------------
    char* ws = (char*)d_ws;
    size_t off = 0;
    auto alloc = [&](size_t bytes) -> void* {
        void* p = ws + off;
        off = (off + bytes + 255) & ~(size_t)255;
        return p;
    };
    const int NR = NB * LMAX;                      // 768 rows for A/V
    int*   lwA   = (int*)  alloc(NR * 4);
    int*   lbA   = (int*)  alloc(NR * 4);
    int*   lwV   = (int*)  alloc(NR * 4);
    int*   lbV   = (int*)  alloc(NR * 4);
    int*   valid = (int*)  alloc(NR * 4);
    float* SwA   = (float*)alloc(NR * 4);
    float* SbA   = (float*)alloc(NR * 4);
    float* SwV   = (float*)alloc(NR * 4);
    float* SbV   = (float*)alloc(NR * 4);
    float* SwI   = (float*)alloc(NB * 4);
    float* SbI   = (float*)alloc(NB * 4);
    float* SwO   = (float*)alloc(NB * 4);
    float* SbO   = (float*)alloc(NB * 4);
    float* embAV = (float*)alloc((size_t)NR * ENCIN * 4);  // reused for A then V
    float* embI  = (float*)alloc((size_t)NB * ENCIN * 4);
    float* embO  = (float*)alloc((size_t)NB * ENCIN * 4);
    float* encI  = (float*)alloc((size_t)NB * HID * 4);
    float* encO  = (float*)alloc((size_t)NB * HID * 4);
    float* encA  = (float*)alloc((size_t)NR * HID * 4);
    float* merged= (float*)alloc((size_t)NR * HID * 4);
    float* h0b   = (float*)alloc((size_t)NB * HR * 4);
    float* h1b   = (float*)alloc((size_t)NB * HR * 4);
    float* hb[2] = { h0b, h1b };

    // ---- stage 1: scrambled lens -------------------------------------------
    k_lens<<<1, 128, 0, stream>>>(gnn, Awp, Abp, Vwp, Vbp, lwA, lbA, lwV, lbV, valid);

    // ---- stage 2: masked row sums ------------------------------------------
    k_rowsum<<<3, 256, 0, stream>>>(A_weight, lwA, PW,   NR, SwA);
    k_rowsum<<<3, 256, 0, stream>>>(A_bias,   lbA, PB,   NR, SbA);
    k_rowsum<<<3, 256, 0, stream>>>(V_weight, lwV, PW,   NR, SwV);
    k_rowsum<<<3, 256, 0, stream>>>(V_bias,   lbV, PB,   NR, SbV);
    k_rowsum<<<1, 128, 0, stream>>>(in_weight,  Iwp, POWW, NB, SwI);
    k_rowsum<<<1, 128, 0, stream>>>(in_bias,    Ibp, POB,  NB, SbI);
    k_rowsum<<<1, 128, 0, stream>>>(out_weight, Owp, POWW, NB, SwO);
    k_rowsum<<<1, 128, 0, stream>>>(out_bias,   Obp, POB,  NB, SbO);

    // ---- stage 3: embeddings + encoders ------------------------------------
    const int gbI = (NB * ENCIN + 255) / 256;
    const int gbR = (NR * ENCIN + 255) / 256;
    // in consumer (small M -> 1x1 tiles for max wave parallelism)
    k_emb<<<gbI, 256, 0, stream>>>(SwI, Iwp, SbI, Ibp, P[P_IN_WT_W], P[P_IN_WT_B],
                                   P[P_IN_BT_W], P[P_IN_BT_B], embI, NB);
    launch_gemm(stream, false, embI, ENCIN, ENCIN, P[P_IN_ENC_W], ENCIN,
                nullptr, 0, 0, nullptr, 0, P[P_IN_ENC_B], nullptr, nullptr,
                nullptr, 0, nullptr, encI, HID, NB, HID, /*relu*/1);
    // out consumer
    k_emb<<<gbI, 256, 0, stream>>>(SwO, Owp, SbO, Obp, P[P_OUT_WT_W], P[P_OUT_WT_B],
                                   P[P_OUT_BT_W], P[P_OUT_BT_B], embO, NB);
    launch_gemm(stream, false, embO, ENCIN, ENCIN, P[P_OUT_ENC_W], ENCIN,
                nullptr, 0, 0, nullptr, 0, P[P_OUT_ENC_B], nullptr, nullptr,
                nullptr, 0, nullptr, encO, HID, NB, HID, 1);
    // A consumer -> encA   (big GEMM: 2x2 register blocking)
    k_emb<<<gbR, 256, 0, stream>>>(SwA, lwA, SbA, lbA, P[P_A_WT_W], P[P_A_WT_B],
                                   P[P_A_BT_W], P[P_A_BT_B], embAV, NR);
    launch_gemm(stream, true, embAV, ENCIN, ENCIN, P[P_A_ENC_W], ENCIN,
                nullptr, 0, 0, nullptr, 0, P[P_A_ENC_B], nullptr, nullptr,
                nullptr, 0, nullptr, encA, HID, NR, HID, 1);
    // V consumer -> merged = relu(encV) + encA   (emb buffer reused)
    k_emb<<<gbR, 256, 0, stream>>>(SwV, lwV, SbV, lbV, P[P_V_WT_W], P[P_V_WT_B],
                                   P[P_V_BT_W], P[P_V_BT_B], embAV, NR);
    launch_gemm(stream, true, embAV, ENCIN, ENCIN, P[P_V_ENC_W], ENCIN,
                nullptr, 0, 0, nullptr, 0, P[P_V_ENC_B], nullptr, encA,
                nullptr, 0, nullptr, merged, HID, NR, HID, 1);

    // ---- stage 4: h0 = encI @ si_w^T + si_b --------------------------------
    launch_gemm(stream, false, encI, HID, HID, P[P_SI_W], HID,
                nullptr, 0, 0, nullptr, 0, P[P_SI_B], nullptr, nullptr,
                nullptr, 0, nullptr, hb[0], HR, NB, HR, 0);

    // ---- stage 5: RNN scan over LMAX steps ---------------------------------
    for (int l = 0; l < LMAX; ++l) {
        launch_gemm(stream, false,
                    merged + (size_t)l * HID, LMAX * HID, HID, P[P_RNN_WIH], HID,
                    hb[l & 1], HR, HR, P[P_RNN_WHH], HR,
                    P[P_RNN_BIH], P[P_RNN_BHH], nullptr,
                    valid + l, LMAX, hb[l & 1],
                    hb[(l + 1) & 1], HR, NB, HR, /*tanh*/2);
    }
    float* hfin = hb[LMAX & 1];

    // ---- stage 6: out = [h_fin, encO] @ outp_w^T + outp_b ------------------
    launch_gemm(stream, false, hfin, HR, HR, P[P_OUTP_W], HR + HID,
                encO, HID, HID, P[P_OUTP_W] + HR, HR + HID,
                P[P_OUTP_B], nullptr, nullptr, nullptr, 0, nullptr,
                (float*)d_out, OUTD, NB, OUTD, 0);
}